// BranchedLinear_90048284328399
// MI455X (gfx1250) — compile-verified
//
#include <hip/hip_runtime.h>

typedef float v2f __attribute__((ext_vector_type(2)));
typedef float v8f __attribute__((ext_vector_type(8)));

#define NUM_BRANCHES 64
#define IN_F 32
#define OUT_F 32
#define BATCH 16384

#define ROWS_PER_WAVE 16
#define WAVES_PER_BLOCK 8
#define ROWS_PER_BLOCK (ROWS_PER_WAVE * WAVES_PER_BLOCK)   // 128

// out[n, b*32 + o] = sum_k x[n, b*32 + k] * W[b, k, o] + bias[b, o]
// One wave32 computes a 16(row) x 32(out-col, one branch) tile with
// V_WMMA_F32_16X16X4_F32, chaining 8 K-steps into the f32 accumulators.
__global__ __launch_bounds__(256, 1) void branched_linear_wmma(
    const float* __restrict__ x,     // [BATCH, 64*32]
    const float* __restrict__ W,     // [64, 32, 32]
    const float* __restrict__ bias,  // [64, 32]
    float* __restrict__ out)         // [BATCH, 64*32]
{
    const int lane   = threadIdx.x & 31;
    const int wave   = threadIdx.x >> 5;
    const int branch = blockIdx.x;                              // 0..63
    const int row0   = blockIdx.y * ROWS_PER_BLOCK + wave * ROWS_PER_WAVE;

    const int m    = lane & 15;          // M (A) / N (B,C,D) index within tile
    const int koff = (lane >> 4) << 1;   // K sub-offset: 0 for lanes 0-15, 2 for 16-31

    const size_t row_stride = (size_t)NUM_BRANCHES * IN_F;      // 2048 floats

    // ---- A fragments: x[row0..row0+15, branch*32 .. +32) ----
    // f32 A layout (16x4 per step): lane m holds A[m][koff], A[m][koff+1].
    const float* xrow = x + (size_t)(row0 + m) * row_stride + branch * IN_F;
    v2f a[8];
#pragma unroll
    for (int k = 0; k < 8; ++k)
        a[k] = *reinterpret_cast<const v2f*>(xrow + 4 * k + koff);

    // ---- B fragments: W[branch] is [IN=32][OUT=32] row-major ----
    // f32 B layout (4x16 per step): lane holds B[koff][n], B[koff+1][n], n = nt*16+m.
    const float* Wb = W + (size_t)branch * IN_F * OUT_F;
    v2f bf[2][8];
#pragma unroll
    for (int nt = 0; nt < 2; ++nt) {
#pragma unroll
        for (int k = 0; k < 8; ++k) {
            const int kr = 4 * k + koff;
            v2f t;
            t.x = Wb[(size_t)kr * OUT_F + nt * 16 + m];
            t.y = Wb[(size_t)(kr + 1) * OUT_F + nt * 16 + m];
            bf[nt][k] = t;
        }
    }

    // ---- C init = bias (each lane's column n is fixed across all 8 C VGPRs) ----
    const float bv0 = bias[branch * OUT_F + m];
    const float bv1 = bias[branch * OUT_F + 16 + m];
    v8f c0, c1;
#pragma unroll
    for (int r = 0; r < 8; ++r) { c0[r] = bv0; c1[r] = bv1; }

    // ---- 8 chained K-steps x 2 N-tiles of v_wmma_f32_16x16x4_f32 ----
#pragma unroll
    for (int k = 0; k < 8; ++k) {
        c0 = __builtin_amdgcn_wmma_f32_16x16x4_f32(
                 false, a[k], false, bf[0][k], (short)0, c0, false, false);
        c1 = __builtin_amdgcn_wmma_f32_16x16x4_f32(
                 false, a[k], false, bf[1][k], (short)0, c1, false, false);
    }

    // ---- store: D layout -> lane writes row (row0 + r + 8*(lane/16)), col nt*16+m ----
    const int mrow = (lane >> 4) * 8;
    float* orow = out + (size_t)(row0 + mrow) * row_stride + branch * OUT_F + m;
#pragma unroll
    for (int r = 0; r < 8; ++r) {
        orow[(size_t)r * row_stride]      = c0[r];
        orow[(size_t)r * row_stride + 16] = c1[r];
    }
}

extern "C" void kernel_launch(void* const* d_in, const int* in_sizes, int n_in,
                              void* d_out, int out_size, void* d_ws, size_t ws_size,
                              hipStream_t stream) {
    (void)in_sizes; (void)n_in; (void)d_ws; (void)ws_size; (void)out_size;
    const float* x    = (const float*)d_in[0];   // [16384, 2048]
    const float* W    = (const float*)d_in[1];   // [64, 32, 32]
    const float* bias = (const float*)d_in[2];   // [64, 32]
    float* out        = (float*)d_out;           // [16384, 2048]

    dim3 grid(NUM_BRANCHES, BATCH / ROWS_PER_BLOCK);  // (64, 128)
    dim3 block(WAVES_PER_BLOCK * 32);                 // 256 threads = 8 waves
    branched_linear_wmma<<<grid, block, 0, stream>>>(x, W, bias, out);
}